// TabNetEncoder_17386027614502
// MI455X (gfx1250) — compile-verified
//
#include <hip/hip_runtime.h>
#include <math.h>

// MI455X / gfx1250, wave32. f32 WMMA (16x16x4) for reference-grade precision.

typedef __attribute__((ext_vector_type(2))) float v2f;
typedef __attribute__((ext_vector_type(8))) float v8f;

#define B_TOTAL 32768
#define D_IN    128
#define HH      384
#define OUTD    192
#define ND      64
#define NSTEPS  3
#define LDA     196      // padded LDS stride for A-operand buffer (bank-conflict free)
#define EPSF    1e-5f
#define GAMMA_F 1.3f

// ws layout (floats)
#define WS_S1    0        // 64*128 partial sums
#define WS_S2    8192     // 64*128 partial sumsq
#define WS_SCALE 16384    // 128
#define WS_BIAS  16512    // 128
#define WS_MLOSS 16640    // 256 per-chunk partials

__device__ __forceinline__ float sigmoidf_(float x) { return 1.0f / (1.0f + expf(-x)); }

// ---------------------------------------------------------------------------
// Kernel A: full-batch BN partial sums (64 blocks x 512 rows each)
// ---------------------------------------------------------------------------
__global__ void bn_stats_kernel(const float* __restrict__ x, float* __restrict__ ws) {
    __shared__ float sh[256];
    const int tid = threadIdx.x;
    const int b   = blockIdx.x;            // 0..63
    const int col = tid & 127;
    const int grp = tid >> 7;              // 0/1
    const float* xp = x + ((size_t)(b * 512 + grp * 256)) * D_IN + col;
    float s1 = 0.f, s2 = 0.f;
    for (int i = 0; i < 256; ++i) { float v = xp[(size_t)i * D_IN]; s1 += v; s2 += v * v; }
    sh[tid] = s1; __syncthreads();
    if (grp == 0) ws[WS_S1 + b * 128 + col] = s1 + sh[tid + 128];
    __syncthreads();
    sh[tid] = s2; __syncthreads();
    if (grp == 0) ws[WS_S2 + b * 128 + col] = s2 + sh[tid + 128];
}

// ---------------------------------------------------------------------------
// Kernel A2: finalize BN -> scale/bias per feature (deterministic fixed order)
// ---------------------------------------------------------------------------
__global__ void bn_final_kernel(float* __restrict__ ws,
                                const float* __restrict__ gamma,
                                const float* __restrict__ beta) {
    const int c = threadIdx.x;  // 128 threads
    float s1 = 0.f, s2 = 0.f;
    for (int b = 0; b < 64; ++b) { s1 += ws[WS_S1 + b * 128 + c]; s2 += ws[WS_S2 + b * 128 + c]; }
    const float mean = s1 * (1.0f / (float)B_TOTAL);
    float var = s2 * (1.0f / (float)B_TOTAL) - mean * mean;
    if (var < 0.f) var = 0.f;
    const float inv = 1.0f / sqrtf(var + EPSF);
    const float sc  = gamma[c] * inv;
    ws[WS_SCALE + c] = sc;
    ws[WS_BIAS + c]  = beta[c] - mean * sc;
}

// ---------------------------------------------------------------------------
// GEMM (A[128xK] @ W[384xK]^T) + GhostBN + GLU, per-wave column-pair tiles.
// MODE 0: write GLU result into sOut (A-buffer, stride LDA), cols 0..191
// MODE 1: dec head: write sigmoid(GLU) to global gOut (row*64 + j), j<64
// MODE 2: att update: sOut=attM: att = 1.3*sigmoid(GLU)*(1-M) in place
// Wave w handles out-tiles t = TBASE + w + 4*p, p in [0,NPAIR)
// ---------------------------------------------------------------------------
template <int K, int NPAIR, int TBASE, int MODE>
__device__ __forceinline__ void gemm_glu(const float* sA,
                                         const float* __restrict__ W,
                                         const float* __restrict__ gamH,
                                         const float* __restrict__ betH,
                                         float* sOut, float* __restrict__ gOut,
                                         int wave, int lane) {
    const int hi = lane >> 4;    // 0/1: which K/row half this lane holds
    const int ln = lane & 15;

    v8f acc[NPAIR][2][8];
#pragma unroll
    for (int p = 0; p < NPAIR; ++p)
#pragma unroll
        for (int h = 0; h < 2; ++h)
#pragma unroll
            for (int rt = 0; rt < 8; ++rt)
#pragma unroll
                for (int e = 0; e < 8; ++e) acc[p][h][rt][e] = 0.f;

    for (int k = 0; k < K; k += 4) {
        v2f bf[NPAIR][2];
#pragma unroll
        for (int p = 0; p < NPAIR; ++p) {
            const int t  = TBASE + wave + 4 * p;
            const int ca = 16 * t + ln;                    // a-column (N index)
            bf[p][0] = *(const v2f*)(W + (size_t)ca * K + k + 2 * hi);
            bf[p][1] = *(const v2f*)(W + (size_t)(ca + 192) * K + k + 2 * hi);
        }
#pragma unroll
        for (int rt = 0; rt < 8; ++rt) {
            v2f a = *(const v2f*)(sA + (rt * 16 + ln) * LDA + k + 2 * hi);
#pragma unroll
            for (int p = 0; p < NPAIR; ++p) {
                acc[p][0][rt] = __builtin_amdgcn_wmma_f32_16x16x4_f32(
                    false, a, false, bf[p][0], (short)0, acc[p][0][rt], false, false);
                acc[p][1][rt] = __builtin_amdgcn_wmma_f32_16x16x4_f32(
                    false, a, false, bf[p][1], (short)0, acc[p][1][rt], false, false);
            }
        }
    }

    // Ghost-BN column stats (column fully inside wave: lanes ln and ln+16)
    float fs[NPAIR][2], fb[NPAIR][2];
#pragma unroll
    for (int p = 0; p < NPAIR; ++p) {
        const int t = TBASE + wave + 4 * p;
#pragma unroll
        for (int h = 0; h < 2; ++h) {
            float s1 = 0.f;
#pragma unroll
            for (int rt = 0; rt < 8; ++rt)
#pragma unroll
                for (int e = 0; e < 8; ++e) s1 += acc[p][h][rt][e];
            s1 += __shfl_xor(s1, 16, 32);
            const float mean = s1 * (1.f / 128.f);
            float s2 = 0.f;
#pragma unroll
            for (int rt = 0; rt < 8; ++rt)
#pragma unroll
                for (int e = 0; e < 8; ++e) {
                    float d = acc[p][h][rt][e] - mean;
                    s2 += d * d;
                }
            s2 += __shfl_xor(s2, 16, 32);
            const float inv = 1.0f / sqrtf(s2 * (1.f / 128.f) + EPSF);
            const int col   = 16 * t + ln + (h ? 192 : 0);
            fs[p][h] = gamH[col] * inv;
            fb[p][h] = betH[col] - mean * fs[p][h];
        }
    }

    if (MODE == 0) __syncthreads();  // all waves done reading sA before overwrite

#pragma unroll
    for (int p = 0; p < NPAIR; ++p) {
        const int t = TBASE + wave + 4 * p;
        const int j = 16 * t + ln;  // GLU output column
#pragma unroll
        for (int rt = 0; rt < 8; ++rt) {
#pragma unroll
            for (int e = 0; e < 8; ++e) {
                const int row  = rt * 16 + hi * 8 + e;
                const float av = acc[p][0][rt][e] * fs[p][0] + fb[p][0];
                const float gv = acc[p][1][rt][e] * fs[p][1] + fb[p][1];
                const float gl = av * sigmoidf_(gv);
                if (MODE == 0) {
                    sOut[row * LDA + j] = gl;
                } else if (MODE == 1) {
                    gOut[(size_t)row * ND + j] = sigmoidf_(gl);
                } else {
                    const int ci = row * 128 + (j - 64);
                    sOut[ci] = GAMMA_F * sigmoidf_(gl) * (1.0f - sOut[ci]);
                }
            }
        }
    }
}

// ---------------------------------------------------------------------------
// Main per-chunk kernel: 256 blocks x 128 threads (4 waves), chunk = 128 rows
// ---------------------------------------------------------------------------
__global__ __launch_bounds__(128, 1) void tabnet_main_kernel(
    const float* __restrict__ x,
    const float* __restrict__ eW0, const float* __restrict__ eg0, const float* __restrict__ eb0,
    const float* __restrict__ eW1, const float* __restrict__ eg1, const float* __restrict__ eb1,
    const float* __restrict__ dW0, const float* __restrict__ dg0, const float* __restrict__ db0,
    const float* __restrict__ dW1, const float* __restrict__ dg1, const float* __restrict__ db1,
    const float* __restrict__ ws, float* __restrict__ wsMloss,
    float* __restrict__ out) {
    extern __shared__ float smem[];
    float* sXn  = smem;                 // [128][128] BN-normalized x
    float* sM   = sXn + 16384;          // [128][128] att / M (in place)
    float* sA   = sM + 16384;           // [128][LDA] GEMM A operand / GLU out
    float* sScl = sA + 128 * LDA;       // [128]
    float* sBia = sScl + 128;           // [128]
    float* sMl  = sBia + 128;           // [4] per-wave M_loss partials

    const int tid   = threadIdx.x;
    const int lane  = tid & 31;
    const int wave  = tid >> 5;
    const int chunk = blockIdx.x;

    if (tid < 128) { sScl[tid] = ws[WS_SCALE + tid]; sBia[tid] = ws[WS_BIAS + tid]; }
    if (tid < 4) sMl[tid] = 0.f;
    __syncthreads();

    const float* xc = x + (size_t)chunk * 128 * D_IN;
    for (int i = tid; i < 16384; i += 128) {
        const int c = i & 127;
        sXn[i] = xc[i] * sScl[c] + sBia[c];
        sM[i]  = 1.0f;   // initial attention (prior stays ones in this model)
    }
    __syncthreads();

    for (int s = 0; s < NSTEPS; ++s) {
        // ---- sparsemax per row (wave-cooperative, exact counting method) ----
        float ml = 0.f;
        for (int r = wave * 32; r < wave * 32 + 32; ++r) {
            float* zr = sM + r * 128;
            float z0 = zr[lane], z1 = zr[lane + 32], z2 = zr[lane + 64], z3 = zr[lane + 96];
            float c0 = 0, c1 = 0, c2 = 0, c3 = 0, S0 = 0, S1 = 0, S2 = 0, S3 = 0;
#pragma unroll
            for (int g = 0; g < 4; ++g) {
                const float zz = (g == 0) ? z0 : (g == 1) ? z1 : (g == 2) ? z2 : z3;
                for (int t = 0; t < 32; ++t) {
                    const float v = __shfl(zz, t, 32);
                    if (v >= z0) { c0 += 1.f; S0 += v; }
                    if (v >= z1) { c1 += 1.f; S1 += v; }
                    if (v >= z2) { c2 += 1.f; S2 += v; }
                    if (v >= z3) { c3 += 1.f; S3 += v; }
                }
            }
            float bk = 0.f, bt = 0.f;
            if ((1.f + c0 * z0 > S0) && (c0 > bk)) { bk = c0; bt = (S0 - 1.f) / c0; }
            if ((1.f + c1 * z1 > S1) && (c1 > bk)) { bk = c1; bt = (S1 - 1.f) / c1; }
            if ((1.f + c2 * z2 > S2) && (c2 > bk)) { bk = c2; bt = (S2 - 1.f) / c2; }
            if ((1.f + c3 * z3 > S3) && (c3 > bk)) { bk = c3; bt = (S3 - 1.f) / c3; }
#pragma unroll
            for (int off = 16; off; off >>= 1) {
                const float ok = __shfl_xor(bk, off, 32);
                const float ot = __shfl_xor(bt, off, 32);
                if (ok > bk) { bk = ok; bt = ot; }
            }
            const float m0 = fmaxf(z0 - bt, 0.f), m1 = fmaxf(z1 - bt, 0.f);
            const float m2 = fmaxf(z2 - bt, 0.f), m3 = fmaxf(z3 - bt, 0.f);
            zr[lane] = m0; zr[lane + 32] = m1; zr[lane + 64] = m2; zr[lane + 96] = m3;
            ml += m0 * logf(m0 + 1e-10f) + m1 * logf(m1 + 1e-10f) +
                  m2 * logf(m2 + 1e-10f) + m3 * logf(m3 + 1e-10f);
        }
#pragma unroll
        for (int off = 16; off; off >>= 1) ml += __shfl_xor(ml, off, 32);
        if (lane == 0) sMl[wave] += ml;
        __syncthreads();

        // ---- masked_x = M * xn into A-buffer ----
        for (int i = tid; i < 16384; i += 128) {
            const int r = i >> 7, c = i & 127;
            sA[r * LDA + c] = sM[i] * sXn[i];
        }
        __syncthreads();

        // ---- DEC chain (produces this step's output) ----
        gemm_glu<128, 3, 0, 0>(sA, dW0 + (size_t)s * HH * D_IN, dg0 + s * HH, db0 + s * HH,
                               sA, nullptr, wave, lane);
        __syncthreads();
        gemm_glu<192, 1, 0, 1>(sA, dW1 + (size_t)s * HH * OUTD, dg1 + s * HH, db1 + s * HH,
                               nullptr, out + ((size_t)s * B_TOTAL + (size_t)chunk * 128) * ND,
                               wave, lane);
        __syncthreads();

        // ---- ENC chain (only needed to update attention; dead at last step) ----
        if (s < NSTEPS - 1) {
            for (int i = tid; i < 16384; i += 128) {
                const int r = i >> 7, c = i & 127;
                sA[r * LDA + c] = sM[i] * sXn[i];
            }
            __syncthreads();
            gemm_glu<128, 3, 0, 0>(sA, eW0 + (size_t)s * HH * D_IN, eg0 + s * HH, eb0 + s * HH,
                                   sA, nullptr, wave, lane);
            __syncthreads();
            gemm_glu<192, 2, 4, 2>(sA, eW1 + (size_t)s * HH * OUTD, eg1 + s * HH, eb1 + s * HH,
                                   sM, nullptr, wave, lane);
        }
        __syncthreads();
    }
    if (tid == 0) wsMloss[chunk] = sMl[0] + sMl[1] + sMl[2] + sMl[3];
}

// ---------------------------------------------------------------------------
// Kernel C: deterministic fixed-order M_loss reduction
// ---------------------------------------------------------------------------
__global__ void mloss_final_kernel(const float* __restrict__ ws, float* __restrict__ outScalar) {
    if (threadIdx.x == 0) {
        float s = 0.f;
        for (int i = 0; i < 256; ++i) s += ws[WS_MLOSS + i];
        outScalar[0] = s / ((float)B_TOTAL * (float)NSTEPS);
    }
}

// ---------------------------------------------------------------------------
extern "C" void kernel_launch(void* const* d_in, const int* in_sizes, int n_in,
                              void* d_out, int out_size, void* d_ws, size_t ws_size,
                              hipStream_t stream) {
    (void)in_sizes; (void)n_in; (void)out_size; (void)ws_size;
    const float* x   = (const float*)d_in[0];
    const float* ig  = (const float*)d_in[1];
    const float* ib  = (const float*)d_in[2];
    const float* eW0 = (const float*)d_in[3];
    const float* eg0 = (const float*)d_in[4];
    const float* eb0 = (const float*)d_in[5];
    const float* eW1 = (const float*)d_in[6];
    const float* eg1 = (const float*)d_in[7];
    const float* eb1 = (const float*)d_in[8];
    const float* dW0 = (const float*)d_in[9];
    const float* dg0 = (const float*)d_in[10];
    const float* db0 = (const float*)d_in[11];
    const float* dW1 = (const float*)d_in[12];
    const float* dg1 = (const float*)d_in[13];
    const float* db1 = (const float*)d_in[14];
    float* ws  = (float*)d_ws;
    float* out = (float*)d_out;

    bn_stats_kernel<<<64, 256, 0, stream>>>(x, ws);
    bn_final_kernel<<<1, 128, 0, stream>>>(ws, ig, ib);

    const size_t smemBytes = (size_t)(16384 + 16384 + 128 * LDA + 128 + 128 + 4) * sizeof(float);
    (void)hipFuncSetAttribute((const void*)tabnet_main_kernel,
                              hipFuncAttributeMaxDynamicSharedMemorySize, (int)smemBytes);
    tabnet_main_kernel<<<256, 128, smemBytes, stream>>>(
        x, eW0, eg0, eb0, eW1, eg1, eb1, dW0, dg0, db0, dW1, dg1, db1,
        ws, ws + WS_MLOSS, out);

    mloss_final_kernel<<<1, 32, 0, stream>>>(ws, out + (size_t)NSTEPS * B_TOTAL * ND);
}